// GNN_12876311953572
// MI455X (gfx1250) — compile-verified
//
#include <hip/hip_runtime.h>

#define N_NODES 100000
#define N_EDGES 800000
#define HIDDEN  128

typedef __attribute__((ext_vector_type(2))) float v2f;
typedef __attribute__((ext_vector_type(8))) float v8f;

// ---------------------------------------------------------------------------
// Degree count: one thread per edge, f32 atomic increment at dst.
// ---------------------------------------------------------------------------
__global__ void gnn_count_edges(const int* __restrict__ ei,
                                float* __restrict__ cnt, int E) {
    int e = blockIdx.x * blockDim.x + threadIdx.x;
    if (e < E) {
        int dst = ei[E + e];                 // edge_index row 1
        atomicAdd(&cnt[dst], 1.0f);
    }
}

// ---------------------------------------------------------------------------
// Scatter-add of source features to agg[dst]. One thread handles a 4-float
// chunk of one edge: float4 gather (coalesced within the 32 threads of an
// edge) + 4 global_atomic_add_f32. Total 25.6M threads -> HBM/atomic bound.
// ---------------------------------------------------------------------------
__global__ void gnn_scatter(const float* __restrict__ feat,
                            const int* __restrict__ ei,
                            float* __restrict__ agg, int E) {
    int idx = blockIdx.x * blockDim.x + threadIdx.x;   // E*32 threads
    int e  = idx >> 5;
    int f4 = (idx & 31) << 2;
    if (e < E) {
        int src = ei[e];
        int dst = ei[E + e];
        const float4 v = *(const float4*)(feat + (size_t)src * HIDDEN + f4);
        float* a = agg + (size_t)dst * HIDDEN + f4;
        atomicAdd(a + 0, v.x);
        atomicAdd(a + 1, v.y);
        atomicAdd(a + 2, v.z);
        atomicAdd(a + 3, v.w);
    }
}

// ---------------------------------------------------------------------------
// Fused SAGE layer: out[16 x 128] = mean @ W_l + x @ W_r + b  (+ReLU).
// Block = 128 threads (4 wave32). LDS-staged x/mean tiles (stride 132 to
// avoid bank conflicts). Each wave computes two 16x16 tiles with
// V_WMMA_F32_16X16X4_F32, chaining C through 64 WMMAs.
// ---------------------------------------------------------------------------
__global__ __launch_bounds__(128)
void gnn_sage_layer(const float* __restrict__ X,
                    const float* __restrict__ agg,
                    const float* __restrict__ cnt,
                    const float* __restrict__ Wl,
                    const float* __restrict__ Wr,
                    const float* __restrict__ bias,
                    float* __restrict__ out,
                    int do_relu) {
    constexpr int LD = HIDDEN + 4;           // 132-float row stride in LDS
    __shared__ float Xt[16 * LD];
    __shared__ float Mt[16 * LD];

    const int tid  = threadIdx.x;
    const int row0 = blockIdx.x * 16;        // N_NODES % 16 == 0, no guard

    // Coalesced tile load; mean computed on the fly.
    #pragma unroll
    for (int i = 0; i < 16; ++i) {
        float rc = 1.0f / fmaxf(cnt[row0 + i], 1.0f);
        Xt[i * LD + tid] = X[(size_t)(row0 + i) * HIDDEN + tid];
        Mt[i * LD + tid] = agg[(size_t)(row0 + i) * HIDDEN + tid] * rc;
    }
    __syncthreads();

    const int wave  = tid >> 5;
    const int lane  = tid & 31;
    const int m     = lane & 15;             // M (for A) / N (for B,C,D)
    const int khalf = (lane >> 4) << 1;      // lanes 16-31 carry K+2,K+3

    #pragma unroll
    for (int t = 0; t < 2; ++t) {
        const int col0 = (wave * 2 + t) * 16;

        // C init with bias (column-broadcast: all 8 C VGPRs identical).
        const float bv = bias[col0 + m];
        v8f c = {bv, bv, bv, bv, bv, bv, bv, bv};

        #pragma unroll 4
        for (int k0 = 0; k0 < HIDDEN; k0 += 4) {
            const int kb = k0 + khalf;
            v2f a, b;

            // mean @ W_l
            a.x = Mt[m * LD + kb];
            a.y = Mt[m * LD + kb + 1];
            b.x = Wl[kb * HIDDEN + col0 + m];
            b.y = Wl[(kb + 1) * HIDDEN + col0 + m];
            c = __builtin_amdgcn_wmma_f32_16x16x4_f32(
                    false, a, false, b, (short)0, c, false, false);

            // x @ W_r
            a.x = Xt[m * LD + kb];
            a.y = Xt[m * LD + kb + 1];
            b.x = Wr[kb * HIDDEN + col0 + m];
            b.y = Wr[(kb + 1) * HIDDEN + col0 + m];
            c = __builtin_amdgcn_wmma_f32_16x16x4_f32(
                    false, a, false, b, (short)0, c, false, false);
        }

        // D layout: VGPR r -> M=r (lanes 0-15) / M=r+8 (lanes 16-31).
        const int rbase = row0 + ((lane >> 4) << 3);
        #pragma unroll
        for (int r = 0; r < 8; ++r) {
            float v = c[r];
            if (do_relu) v = fmaxf(v, 0.0f);
            out[(size_t)(rbase + r) * HIDDEN + col0 + m] = v;
        }
    }
}

// ---------------------------------------------------------------------------
extern "C" void kernel_launch(void* const* d_in, const int* in_sizes, int n_in,
                              void* d_out, int out_size, void* d_ws, size_t ws_size,
                              hipStream_t stream) {
    const float* x   = (const float*)d_in[0];
    const int*   ei  = (const int*)d_in[1];
    const float* Wl1 = (const float*)d_in[2];
    const float* Wr1 = (const float*)d_in[3];
    const float* b1  = (const float*)d_in[4];
    const float* Wl2 = (const float*)d_in[5];
    const float* Wr2 = (const float*)d_in[6];
    const float* b2  = (const float*)d_in[7];
    float* out = (float*)d_out;

    char* ws = (char*)d_ws;
    const size_t featBytes = (size_t)N_NODES * HIDDEN * sizeof(float);
    float* agg = (float*)(ws);
    float* h   = (float*)(ws + featBytes);
    float* cnt = (float*)(ws + 2 * featBytes);

    hipMemsetAsync(agg, 0, featBytes, stream);
    hipMemsetAsync(cnt, 0, (size_t)N_NODES * sizeof(float), stream);

    const int scatterBlocks = (N_EDGES * 32) / 256;        // 100000
    gnn_count_edges<<<(N_EDGES + 255) / 256, 256, 0, stream>>>(ei, cnt, N_EDGES);
    gnn_scatter<<<scatterBlocks, 256, 0, stream>>>(x, ei, agg, N_EDGES);

    gnn_sage_layer<<<N_NODES / 16, 128, 0, stream>>>(x, agg, cnt,
                                                     Wl1, Wr1, b1, h, 1);

    hipMemsetAsync(agg, 0, featBytes, stream);
    gnn_scatter<<<scatterBlocks, 256, 0, stream>>>(h, ei, agg, N_EDGES);

    gnn_sage_layer<<<N_NODES / 16, 128, 0, stream>>>(h, agg, cnt,
                                                     Wl2, Wr2, b2, out, 0);
}